// torchShiftAA_82995948028392
// MI455X (gfx1250) — compile-verified
//
#include <hip/hip_runtime.h>
#include <math.h>

#define NN 1000
#define MM 4000
#define RD 8
#define TPI 6.283185307179586f
#define NCHUNK 16
#define NPC 63          // ceil(NN / NCHUNK)
#define NTILES 63       // ceil(NN / 16)
#define MTILES 250      // MM / 16
#define TBYTES (MM * 8) // twiddle table size in bytes (float2 per entry)

typedef __attribute__((ext_vector_type(2))) float v2f;
typedef __attribute__((ext_vector_type(8))) float v8f;

// ---------------------------------------------------------------------------
// Kernel 1: softmax of C_tilde over d (axis 0 of [8,N]) and S_tilde over d
// (axis 1 of [N,8]).  Store both as [n][d] for broadcast-friendly reads.
// ---------------------------------------------------------------------------
__global__ void k_softmax(const float* __restrict__ Ct, const float* __restrict__ St,
                          float* __restrict__ Cs, float* __restrict__ Ss) {
    int n = blockIdx.x * blockDim.x + threadIdx.x;
    if (n >= NN) return;
    float v[RD], mx, sum, inv;

    mx = -1e30f;
    #pragma unroll
    for (int d = 0; d < RD; ++d) { v[d] = Ct[d * NN + n]; mx = fmaxf(mx, v[d]); }
    sum = 0.0f;
    #pragma unroll
    for (int d = 0; d < RD; ++d) { v[d] = __expf(v[d] - mx); sum += v[d]; }
    inv = 1.0f / sum;
    #pragma unroll
    for (int d = 0; d < RD; ++d) Cs[n * RD + d] = v[d] * inv;

    mx = -1e30f;
    #pragma unroll
    for (int d = 0; d < RD; ++d) { v[d] = St[n * RD + d]; mx = fmaxf(mx, v[d]); }
    sum = 0.0f;
    #pragma unroll
    for (int d = 0; d < RD; ++d) { v[d] = __expf(v[d] - mx); sum += v[d]; }
    inv = 1.0f / sum;
    #pragma unroll
    for (int d = 0; d < RD; ++d) Ss[n * RD + d] = v[d] * inv;
}

// ---------------------------------------------------------------------------
// Kernel 2: per-(n,d) phase tables.  theta = 2*pi*1000*tanh(tau_tilde)/M.
// E[nd][j] = exp(i*theta*j)        for j in [0,32)
// E[nd][32+k] = exp(i*theta*32*k)  for k in [0,125)
// so exp(i*theta*m) = E[32 + (m>>5)] * E[m&31].   ~1.26M sincos total
// (vs 32e6 if evaluated per (n,d,m)).
// ---------------------------------------------------------------------------
__global__ void k_tables(const float* __restrict__ taut, float* __restrict__ E) {
    int gid = blockIdx.x * blockDim.x + threadIdx.x;
    if (gid >= NN * RD * 157) return;
    int j  = gid % 157;
    int nd = gid / 157;
    float th  = TPI * tanhf(taut[nd]) * 1000.0f / (float)MM;
    float ang = (j < 32) ? th * (float)j : th * 32.0f * (float)(j - 32);
    float s, c;
    __sincosf(ang, &s, &c);
    E[gid * 2]     = c;
    E[gid * 2 + 1] = s;
}

// ---------------------------------------------------------------------------
// Kernel 3: DFT as f32 WMMA GEMM.  X_F[n,m] = sum_k X[n,k] * exp(-2*pi*i*k*m/M)
// Twiddle W[k,m] = T[(k*m) mod M] from a 4000-entry LDS table (32 KB of the
// 320 KB WGP pool).  Output tile 16(n) x 16(m) per wave, K streamed in
// chunks of 4.  Two running byte-offsets into the LDS table advance by
// (4m mod M)*8 each chunk (one add + cond-subtract each); no per-chunk
// multiplies, no modulo, no shift.
// A frag (16x4 f32): lane%16 = row n, K = vgpr + 2*(lane>>4)  (per ISA layout)
// B frag (4x16):     lane%16 = col m, K = vgpr + 2*(lane>>4)  (mirror of A)
// D frag (16x16):    row = vgpr + 8*(lane>>4), col = lane%16
// Tail rows (n>=1000) read a clamped valid row; their D rows are never
// stored, and since D row r depends only on A row r, no masking is needed.
// ---------------------------------------------------------------------------
__global__ void __launch_bounds__(256) k_dft(const float* __restrict__ X,
                                             float* __restrict__ XF) {
    __shared__ float T[MM * 2];
    int tid = threadIdx.x;
    for (int j = tid; j < MM; j += 256) {
        float s, c;
        __sincosf(-TPI * (float)j / (float)MM, &s, &c);
        T[2 * j]     = c;
        T[2 * j + 1] = s;
    }
    __syncthreads();

    int wave  = tid >> 5;
    int lane  = tid & 31;
    int mtile = blockIdx.y * 8 + wave;
    if (mtile >= MTILES) return;                // wave-uniform exit

    int nbase = blockIdx.x * 16;
    int col   = lane & 15;
    int hi    = lane >> 4;
    int m     = mtile * 16 + col;

    int arow = nbase + col;
    const float* Ap = X + (size_t)((arow < NN) ? arow : (NN - 1)) * MM;

    const char* Tb = (const char*)T;
    int stepb = ((4 * m) % MM) * 8;             // byte step per K-chunk
    int b0 = ((2 * hi * m) % MM) * 8;           // byte offset for lane's K0
    int b1 = b0 + m * 8; if (b1 >= TBYTES) b1 -= TBYTES;  // K0+1

    v8f accR = {};
    v8f accI = {};

    #pragma unroll 4
    for (int kb = 0; kb < MM / 4; ++kb) {
        v2f a = *(const v2f*)(Ap + kb * 4 + 2 * hi);   // X[n][kk], X[n][kk+1]

        float2 w0 = *(const float2*)(Tb + b0);         // ds_load_b64
        float2 w1 = *(const float2*)(Tb + b1);

        v2f bc; bc.x = w0.x; bc.y = w1.x;       // cos(-phi) plane
        v2f bs; bs.x = w0.y; bs.y = w1.y;       // sin(-phi) plane

        accR = __builtin_amdgcn_wmma_f32_16x16x4_f32(false, a, false, bc,
                                                     (short)0, accR, false, false);
        accI = __builtin_amdgcn_wmma_f32_16x16x4_f32(false, a, false, bs,
                                                     (short)0, accI, false, false);

        b0 += stepb;  if (b0 >= TBYTES) b0 -= TBYTES;
        b1 += stepb;  if (b1 >= TBYTES) b1 -= TBYTES;
    }

    // T holds (cos(-phi), sin(-phi)), so X_F = (accR, accI) directly.
    #pragma unroll
    for (int v = 0; v < 8; ++v) {
        int row = nbase + v + 8 * hi;
        if (row < NN) {
            float2 o; o.x = accR[v]; o.y = accI[v];
            *(float2*)(XF + ((size_t)row * MM + m) * 2) = o;
        }
    }
}

// ---------------------------------------------------------------------------
// Kernel 4: partial A[d,m] = sum_{n in chunk} C[d,n] * X_F[n,m] * exp(+i*theta*m)
// Grid: (m-tiles, d, n-chunks).  Deterministic tree reduction via Apart.
// ---------------------------------------------------------------------------
__global__ void k_align(const float* __restrict__ XF, const float* __restrict__ Cs,
                        const float* __restrict__ E, float* __restrict__ Apart) {
    int m = blockIdx.x * blockDim.x + threadIdx.x;
    if (m >= MM) return;
    int d  = blockIdx.y;
    int nc = blockIdx.z;
    int n0 = nc * NPC;
    int n1 = n0 + NPC; if (n1 > NN) n1 = NN;

    int jlo = (m & 31) * 2;
    int jhi = (32 + (m >> 5)) * 2;

    float ar = 0.0f, ai = 0.0f;
    for (int n = n0; n < n1; ++n) {
        int nd = n * RD + d;
        float cdn = Cs[nd];                               // broadcast across lanes
        const float* Ep = E + (size_t)nd * 157 * 2;
        float2 e1 = *(const float2*)(Ep + jlo);
        float2 e3 = *(const float2*)(Ep + jhi);
        float wr = e3.x * e1.x - e3.y * e1.y;             // exp(+i*theta*m)
        float wi = e3.x * e1.y + e3.y * e1.x;
        float2 xf = *(const float2*)(XF + ((size_t)n * MM + m) * 2);
        float tr = xf.x * wr - xf.y * wi;
        float ti = xf.x * wi + xf.y * wr;
        ar = fmaf(cdn, tr, ar);
        ai = fmaf(cdn, ti, ai);
    }
    float2 o; o.x = ar; o.y = ai;
    *(float2*)(Apart + (((size_t)nc * RD + d) * MM + m) * 2) = o;
}

// ---------------------------------------------------------------------------
// Kernel 5: reduce partials over n-chunks (fixed order -> deterministic).
// ---------------------------------------------------------------------------
__global__ void k_reduce(const float* __restrict__ Apart, float* __restrict__ A) {
    int gid = blockIdx.x * blockDim.x + threadIdx.x;    // gid = d*MM + m
    if (gid >= RD * MM) return;
    float ar = 0.0f, ai = 0.0f;
    #pragma unroll
    for (int nc = 0; nc < NCHUNK; ++nc) {
        float2 p = *(const float2*)(Apart + ((size_t)nc * RD * MM + gid) * 2);
        ar += p.x; ai += p.y;
    }
    float2 o; o.x = ar; o.y = ai;
    *(float2*)(A + (size_t)gid * 2) = o;
}

// ---------------------------------------------------------------------------
// Kernel 6: out[n,m] = X[n,m] - sum_d S[n,d] * A[d,m] * exp(-i*theta[n,d]*m)
// Complex64 output, interleaved (re, im).
// ---------------------------------------------------------------------------
__global__ void k_recon(const float* __restrict__ X, const float* __restrict__ A,
                        const float* __restrict__ Ss, const float* __restrict__ E,
                        float* __restrict__ out) {
    int m = blockIdx.x * blockDim.x + threadIdx.x;
    if (m >= MM) return;
    int n = blockIdx.y;
    int jlo = (m & 31) * 2;
    int jhi = (32 + (m >> 5)) * 2;

    float ar = 0.0f, ai = 0.0f;
    #pragma unroll
    for (int d = 0; d < RD; ++d) {
        int nd = n * RD + d;
        float snd = Ss[nd];
        const float* Ep = E + (size_t)nd * 157 * 2;
        float2 e1 = *(const float2*)(Ep + jlo);
        float2 e3 = *(const float2*)(Ep + jhi);
        float wr = e3.x * e1.x - e3.y * e1.y;             // exp(+i*theta*m)
        float wi = e3.x * e1.y + e3.y * e1.x;             // conjugate below
        float2 av = *(const float2*)(A + ((size_t)d * MM + m) * 2);
        float tr = av.x * wr + av.y * wi;                 // av * conj(w)
        float ti = av.y * wr - av.x * wi;
        ar = fmaf(snd, tr, ar);
        ai = fmaf(snd, ti, ai);
    }
    float xv = X[(size_t)n * MM + m];
    float2 o; o.x = xv - ar; o.y = -ai;
    *(float2*)(out + ((size_t)n * MM + m) * 2) = o;
}

// ---------------------------------------------------------------------------
extern "C" void kernel_launch(void* const* d_in, const int* in_sizes, int n_in,
                              void* d_out, int out_size, void* d_ws, size_t ws_size,
                              hipStream_t stream) {
    const float* X  = (const float*)d_in[0];   // [1000,4000]
    const float* Ct = (const float*)d_in[1];   // [8,1000]
    const float* St = (const float*)d_in[2];   // [1000,8]
    const float* Tt = (const float*)d_in[3];   // [1000,8]
    float* out = (float*)d_out;

    char* p = (char*)d_ws;
    auto take = [&](size_t bytes) -> void* {
        void* q = (void*)p;
        p += (bytes + 255) & ~(size_t)255;
        return q;
    };
    float* E     = (float*)take((size_t)NN * RD * 157 * 2 * 4);   // ~10 MB
    float* Cs    = (float*)take((size_t)NN * RD * 4);
    float* Ss    = (float*)take((size_t)NN * RD * 4);
    float* XF    = (float*)take((size_t)NN * MM * 2 * 4);         // 32 MB
    float* Apart = (float*)take((size_t)NCHUNK * RD * MM * 2 * 4);// 4 MB
    float* A     = (float*)take((size_t)RD * MM * 2 * 4);         // 256 KB

    k_softmax<<<dim3((NN + 255) / 256), dim3(256), 0, stream>>>(Ct, St, Cs, Ss);
    k_tables <<<dim3((NN * RD * 157 + 255) / 256), dim3(256), 0, stream>>>(Tt, E);
    k_dft    <<<dim3(NTILES, 32), dim3(256), 0, stream>>>(X, XF);
    k_align  <<<dim3(16, RD, NCHUNK), dim3(256), 0, stream>>>(XF, Cs, E, Apart);
    k_reduce <<<dim3((RD * MM + 255) / 256), dim3(256), 0, stream>>>(Apart, A);
    k_recon  <<<dim3(16, NN), dim3(256), 0, stream>>>(X, A, Ss, E, out);
}